// CheckInEmbedding_532575945120
// MI455X (gfx1250) — compile-verified
//
#include <hip/hip_runtime.h>

// CheckInEmbedding for MI455X (gfx1250):
//   out[n, 0:128]   = leaky_relu(cat(hot[n], reg[n]), 0.2)
//   out[n, 128:256] = user[0:128]  (broadcast)
//
// Pure streaming op (~768 MB traffic, ~33 us floor at 23.3 TB/s).
// Strategy: b128 per-lane moves, non-temporal hints for the streamed
// 256 MB in / 512 MB out, and the gfx1250 async-DMA path
// (global_load_async_to_lds_b128 + s_wait_asynccnt) to stage the 512-byte
// user vector into LDS once per block for the broadcast half.

typedef float v4f __attribute__((ext_vector_type(4)));

__device__ __forceinline__ v4f leaky4(v4f v) {
    v4f r;
    r.x = fmaxf(v.x, 0.2f * v.x);
    r.y = fmaxf(v.y, 0.2f * v.y);
    r.z = fmaxf(v.z, 0.2f * v.z);
    r.w = fmaxf(v.w, 0.2f * v.w);
    return r;
}

// ---------------------------------------------------------------------------
// Kernel 1: poi half. One float4 (16B) per thread.
// Row = 128 floats = 32 float4 chunks; chunk c4 in [0,16) -> hot, [16,32) -> reg.
// Lane-level pointer select compiles to v_cndmask (no branch). NT load + NT
// store: this data is touched exactly once, keep it out of L2.
// ---------------------------------------------------------------------------
__global__ __launch_bounds__(256) void poi_leaky_kernel(
    const v4f* __restrict__ hot,   // [n,64] fp32 == [n,16] v4f
    const v4f* __restrict__ reg,   // [n,64] fp32 == [n,16] v4f
    v4f* __restrict__ out,         // [n,256] fp32 == [n,64] v4f
    int chunks)                    // n * 32
{
    int idx = blockIdx.x * 256 + threadIdx.x;
    if (idx >= chunks) return;
    int row = idx >> 5;          // 32 chunks per row
    int c4  = idx & 31;

    const v4f* base = (c4 < 16) ? hot : reg;
    int cc = c4 & 15;
    v4f v = __builtin_nontemporal_load(base + (size_t)row * 16 + cc);
    v4f r = leaky4(v);
    __builtin_nontemporal_store(r, out + (size_t)row * 64 + c4);
}

// ---------------------------------------------------------------------------
// Kernel 2: user-broadcast half. Stage the 512-byte user vector into LDS via
// the gfx1250 async DMA path (ASYNCcnt-tracked), then every thread reads its
// 16B chunk from LDS (ds_load_b128) and streams it out with an NT store.
// ---------------------------------------------------------------------------
__global__ __launch_bounds__(256) void user_bcast_kernel(
    const float* __restrict__ user,  // [128] fp32
    v4f* __restrict__ out,           // [n,256] fp32 == [n,64] v4f
    int chunks)                      // n * 32
{
    __shared__ v4f lds_user[32];     // 512 bytes

    // Wave 0: async-copy 32 x 16B = 512B global -> LDS, then drain ASYNCcnt.
    if (threadIdx.x < 32) {
        unsigned lane = threadIdx.x;
        // LDS byte offset of this lane's 16B chunk (generic -> AS(3) -> int).
        unsigned lds_off = (unsigned)(unsigned long long)
            (__attribute__((address_space(3))) v4f*)&lds_user[lane];
        const float* gsrc = user + lane * 4;
        asm volatile("global_load_async_to_lds_b128 %0, %1, off"
                     :
                     : "v"(lds_off), "v"(gsrc)
                     : "memory");
        asm volatile("s_wait_asynccnt 0" ::: "memory");
    }
    __syncthreads();

    int idx = blockIdx.x * 256 + threadIdx.x;
    if (idx >= chunks) return;
    int row = idx >> 5;
    int c4  = idx & 31;

    v4f v = lds_user[c4];                                    // ds_load_b128
    __builtin_nontemporal_store(v, out + (size_t)row * 64 + 32 + c4);
}

extern "C" void kernel_launch(void* const* d_in, const int* in_sizes, int n_in,
                              void* d_out, int out_size, void* d_ws, size_t ws_size,
                              hipStream_t stream) {
    (void)n_in; (void)out_size; (void)d_ws; (void)ws_size;

    const float* hot  = (const float*)d_in[0];  // [n,64]
    const float* reg  = (const float*)d_in[1];  // [n,64]
    const float* user = (const float*)d_in[2];  // [128]

    int n      = in_sizes[0] / 64;   // DH == 64
    int chunks = n * 32;             // float4 chunks per 128-float half
    int blocks = (chunks + 255) / 256;

    poi_leaky_kernel<<<blocks, 256, 0, stream>>>(
        (const v4f*)hot, (const v4f*)reg, (v4f*)d_out, chunks);
    user_bcast_kernel<<<blocks, 256, 0, stream>>>(
        user, (v4f*)d_out, chunks);
}